// CA_Block_34368328303292
// MI455X (gfx1250) — compile-verified
//
#include <hip/hip_runtime.h>
#include <math.h>

// Problem constants (match reference)
#define Bq 2
#define Cc 256
#define Dd 32
#define Hh 64
#define Ww 192
#define HW (Hh * Ww)      // 12288
#define CQ 64             // C/4
#define NWAVE 4           // waves per block in WMMA kernels

typedef __attribute__((ext_vector_type(2))) float v2f;
typedef __attribute__((ext_vector_type(8))) float v8f;

// GCC-style vector type matching the async-LDS builtin's parameter type
typedef int v4i_b __attribute__((vector_size(4 * sizeof(int))));
typedef __attribute__((address_space(1))) v4i_b* gas_v4i;
typedef __attribute__((address_space(3))) v4i_b* las_v4i;

#if defined(__has_builtin)
# if __has_builtin(__builtin_amdgcn_wmma_f32_16x16x4_f32)
#  define HAVE_WMMA_F32 1
# endif
# if __has_builtin(__builtin_amdgcn_global_load_async_to_lds_b128)
#  define HAVE_ASYNC_B128 1
# endif
# if __has_builtin(__builtin_amdgcn_s_wait_asynccnt)
#  define HAVE_WAIT_ASYNC 1
# endif
#endif

// ---------------------------------------------------------------------------
// f32 WMMA 16x16x4 (CDNA5).  A: 16x4 (lane=M, vgpr=K, lanes16-31 hold K=2,3)
// B: 4x16 mirrored (lane=N, vgpr=K).  C/D: vgpr i -> M=i+8*(lane>=16), N=lane&15
// ---------------------------------------------------------------------------
__device__ __forceinline__ v8f wmma4(v2f a, v2f b, v8f c) {
#if defined(HAVE_WMMA_F32)
    return __builtin_amdgcn_wmma_f32_16x16x4_f32(false, a, false, b, (short)0, c,
                                                 false, false);
#else
    int lane = (int)(threadIdx.x & 31);
    v8f d = c;
    #pragma unroll
    for (int i = 0; i < 8; ++i) {
        int m = i + 8 * (lane >> 4);
        int n = lane & 15;
        float s = d[i];
        #pragma unroll
        for (int kk = 0; kk < 4; ++kk) {
            float av = __shfl((kk & 1) ? a.y : a.x, (m & 15) + 16 * (kk >> 1), 32);
            float bv = __shfl((kk & 1) ? b.y : b.x, n + 16 * (kk >> 1), 32);
            s = fmaf(av, bv, s);
        }
        d[i] = s;
    }
    return d;
#endif
}

__device__ __forceinline__ void wait_async0() {
#if defined(HAVE_ASYNC_B128)
# if defined(HAVE_WAIT_ASYNC)
    __builtin_amdgcn_s_wait_asynccnt(0);
# else
    asm volatile("s_wait_asynccnt 0x0" ::: "memory");
# endif
#endif
}

// Copy a [CQ x Ww] row block (channel stride = HW floats) into LDS.
// Uses the CDNA5 async global->LDS DMA path when available (ASYNCcnt).
__device__ __forceinline__ void fill_lds_rows(float* dst, const float* src,
                                              int tid, int nthreads) {
#if defined(HAVE_ASYNC_B128)
    const int nvec = (CQ * Ww) / 4;                      // 16B chunks
    for (int i = tid; i < nvec; i += nthreads) {
        int c    = (i * 4) / Ww;
        int wcol = (i * 4) % Ww;
        const float* gp = src + (size_t)c * HW + wcol;   // 16B aligned
        float*       lp = dst + i * 4;                   // 16B aligned
        __builtin_amdgcn_global_load_async_to_lds_b128(
            (gas_v4i)(uintptr_t)gp,
            (las_v4i)(unsigned int)(uintptr_t)lp,
            0, 0);
    }
    wait_async0();
#else
    for (int i = tid; i < CQ * Ww; i += nthreads)
        dst[i] = src[(size_t)(i / Ww) * HW + (i % Ww)];
#endif
}

__device__ __forceinline__ float elu1(float x) {
    return x > 0.f ? x : expm1f(x);
}

__device__ __forceinline__ void warp_params(int d, int w, float sh_unit,
                                            int& x0, int& x1, float& f) {
    float src = (float)w + sh_unit * (float)d;
    src = fminf(fmaxf(src, 0.f), (float)(Ww - 1));
    x0 = (int)floorf(src);
    x1 = min(x0 + 1, Ww - 1);
    f  = src - (float)x0;
}

// ---------------------------------------------------------------------------
// 1x1 conv as WMMA GEMM:  Out[b, m, n] = sum_k W[m,k] * X[k, n]  (+bias, elu)
// X rows 0..K1-1 from X1, rows K1..K1+K2-1 from X2 (concat along channels).
// 4 waves per block; each wave owns one 16x16 tile.
// ---------------------------------------------------------------------------
__global__ void gemm1x1_wmma(const float* __restrict__ Wt, const float* __restrict__ bias,
                             const float* __restrict__ X1, int K1, long x1bs,
                             const float* __restrict__ X2, int K2, long x2bs,
                             float* __restrict__ Out, long obs, int M, int do_elu) {
    const int lane  = (int)threadIdx.x & 31;
    const int wave  = (int)threadIdx.x >> 5;
    const int n0    = (blockIdx.x * NWAVE + wave) * 16;
    const int m0    = blockIdx.y * 16;
    const int b     = blockIdx.z;
    const int K     = K1 + K2;
    const int mrow  = m0 + (lane & 15);
    const int kq    = (lane >> 4) * 2;    // 0 or 2
    const int ncol  = n0 + (lane & 15);

    v8f acc = {};
    for (int k = 0; k < K; k += 4) {
        const int kr0 = k + kq;           // even; kr1 = kr0+1 in same region
        v2f a;
        a.x = Wt[(size_t)mrow * K + kr0];
        a.y = Wt[(size_t)mrow * K + kr0 + 1];
        const float* src;
        size_t roff;
        if (kr0 < K1) { src = X1 + (size_t)b * x1bs; roff = (size_t)kr0 * HW; }
        else          { src = X2 + (size_t)b * x2bs; roff = (size_t)(kr0 - K1) * HW; }
        __builtin_prefetch(src + roff + (size_t)8 * HW + ncol, 0, 1);
        v2f bb;
        bb.x = src[roff + ncol];
        bb.y = src[roff + HW + ncol];
        acc = wmma4(a, bb, acc);
    }
    const int half = lane >> 4;
    #pragma unroll
    for (int i = 0; i < 8; ++i) {
        int m = m0 + i + 8 * half;
        float val = acc[i] + bias[m];
        if (do_elu) val = elu1(val);
        Out[(size_t)b * obs + (size_t)m * HW + ncol] = val;
    }
}

// ---------------------------------------------------------------------------
// 3x3 SAME conv as 9 shifted WMMA GEMMs. X: [B,Cin,H,W], Wt: OIHW.
// mode 0: Out = elu(conv + bias)            (p1 path)
// mode 1: Out = 0.5*(extra + conv + bias)   (p2 path, cost residual)
// 4 waves per block; each wave owns one 16-column tile.
// ---------------------------------------------------------------------------
__global__ void conv3x3_wmma(const float* __restrict__ Wt, const float* __restrict__ bias,
                             const float* __restrict__ X, int Cin,
                             float* __restrict__ Out, int M,
                             const float* __restrict__ extra, int mode) {
    const int lane = (int)threadIdx.x & 31;
    const int wave = (int)threadIdx.x >> 5;
    const int n0   = (blockIdx.x * NWAVE + wave) * 16;
    const int m0   = blockIdx.y * 16;
    const int bh   = blockIdx.z;
    const int b    = bh / Hh;
    const int h    = bh % Hh;
    const int mrow = m0 + (lane & 15);
    const int kq   = (lane >> 4) * 2;
    const int ncol = n0 + (lane & 15);
    if (n0 >= Ww) return;                          // uniform per wave

    v8f acc = {};
    for (int kh = 0; kh < 3; ++kh) {
        int hh = h + kh - 1;
        if (hh < 0 || hh >= Hh) continue;          // zero pad rows (uniform)
        const float* xrow = X + (size_t)b * Cin * HW + (size_t)hh * Ww;
        for (int kw = 0; kw < 3; ++kw) {
            int col = ncol + kw - 1;
            bool ok = (col >= 0) && (col < Ww);    // zero pad cols (per-lane select)
            for (int k = 0; k < Cin; k += 4) {
                int c0 = k + kq;
                v2f a;
                a.x = Wt[(((size_t)mrow * Cin + c0) * 3 + kh) * 3 + kw];
                a.y = Wt[(((size_t)mrow * Cin + c0 + 1) * 3 + kh) * 3 + kw];
                v2f bb;
                bb.x = ok ? xrow[(size_t)c0 * HW + col] : 0.f;
                bb.y = ok ? xrow[(size_t)(c0 + 1) * HW + col] : 0.f;
                acc = wmma4(a, bb, acc);
            }
        }
    }
    const int half = lane >> 4;
    #pragma unroll
    for (int i = 0; i < 8; ++i) {
        int m = m0 + i + 8 * half;
        float val = acc[i] + bias[m];
        size_t oidx = (((size_t)b * M + m) * Hh + h) * Ww + ncol;
        if (mode == 0) val = elu1(val);
        else           val = 0.5f * (extra[oidx] + val);
        Out[oidx] = val;
    }
}

// ---------------------------------------------------------------------------
// cost[b,d,h,w] = (1/8) * sum_c q[c,w] * bilinear(k[c,:], w+shift(d))
// One block per (b,h); k row cached in LDS via async DMA. block=(192,2).
// ---------------------------------------------------------------------------
__global__ void cost_kernel(const float* __restrict__ qbuf,   // [B,128,H,W], q = ch 0..63
                            const float* __restrict__ kbuf,   // [B,64,H,W]
                            const float* __restrict__ directs,
                            const int* __restrict__ imgw,
                            float* __restrict__ cost) {       // [B,32,H,W]
    __shared__ float ks[CQ][Ww];                               // 48 KB
    const int h = blockIdx.x, b = blockIdx.y;
    const int w = (int)threadIdx.x, ty = (int)threadIdx.y;
    const int tid = ty * Ww + w;
    const float* kg = kbuf + (size_t)b * CQ * HW + (size_t)h * Ww;
    fill_lds_rows(&ks[0][0], kg, tid, 2 * Ww);
    __syncthreads();

    const int iw = *imgw;
    const float rs = (iw == 640) ? 1.0f : 640.0f / (float)iw;
    const float sh = 0.01f * rs * directs[b] * (float)(Ww - 1);

    int x0a[16], x1a[16];
    float fa[16];
    #pragma unroll
    for (int i = 0; i < 16; ++i)
        warp_params(ty * 16 + i, w, sh, x0a[i], x1a[i], fa[i]);

    float acc[16];
    #pragma unroll
    for (int i = 0; i < 16; ++i) acc[i] = 0.f;

    const float* qg = qbuf + (size_t)b * 128 * HW + (size_t)h * Ww + w;
    for (int c = 0; c < CQ; ++c) {
        float qc = qg[(size_t)c * HW];
        #pragma unroll
        for (int i = 0; i < 16; ++i) {
            float kv = ks[c][x0a[i]] * (1.f - fa[i]) + ks[c][x1a[i]] * fa[i];
            acc[i] = fmaf(qc, kv, acc[i]);
        }
    }
    #pragma unroll
    for (int i = 0; i < 16; ++i) {
        int d = ty * 16 + i;
        cost[(((size_t)b * Dd + d) * Hh + h) * Ww + w] = acc[i] * 0.125f; // 1/sqrt(64)
    }
}

// ---------------------------------------------------------------------------
// softmax over d, then agg[c,w] = sum_d p[d] * bilinear(v[c,:], w+shift(d)).
// Writes agg into channels 64..127 of qagg (concat buffer for p1 conv).
// ---------------------------------------------------------------------------
__global__ void softmax_agg_kernel(const float* __restrict__ cost,   // [B,32,H,W]
                                   const float* __restrict__ vbuf,   // [B,64,H,W]
                                   const float* __restrict__ directs,
                                   const int* __restrict__ imgw,
                                   float* __restrict__ qagg) {       // [B,128,H,W]
    __shared__ float vs[CQ][Ww];
    const int h = blockIdx.x, b = blockIdx.y;
    const int w = (int)threadIdx.x, ty = (int)threadIdx.y;
    const int tid = ty * Ww + w;
    const float* vg = vbuf + (size_t)b * CQ * HW + (size_t)h * Ww;
    fill_lds_rows(&vs[0][0], vg, tid, 2 * Ww);
    __syncthreads();

    float nc[Dd];
    float mx = -1e30f;
    #pragma unroll
    for (int d = 0; d < Dd; ++d) {
        nc[d] = cost[(((size_t)b * Dd + d) * Hh + h) * Ww + w];
        mx = fmaxf(mx, nc[d]);
    }
    float sum = 0.f;
    #pragma unroll
    for (int d = 0; d < Dd; ++d) { nc[d] = expf(nc[d] - mx); sum += nc[d]; }
    const float inv = 1.f / sum;

    const int iw = *imgw;
    const float rs = (iw == 640) ? 1.0f : 640.0f / (float)iw;
    const float sh = 0.01f * rs * directs[b] * (float)(Ww - 1);

    float acc[32];
    #pragma unroll
    for (int i = 0; i < 32; ++i) acc[i] = 0.f;

    for (int d = 0; d < Dd; ++d) {
        int x0, x1; float f;
        warp_params(d, w, sh, x0, x1, f);
        float p = nc[d] * inv;
        #pragma unroll
        for (int i = 0; i < 32; ++i) {
            int c = ty * 32 + i;
            float vv = vs[c][x0] * (1.f - f) + vs[c][x1] * f;
            acc[i] = fmaf(p, vv, acc[i]);
        }
    }
    #pragma unroll
    for (int i = 0; i < 32; ++i) {
        int c = ty * 32 + i;
        qagg[((size_t)b * 128 + 64 + c) * HW + (size_t)h * Ww + w] = acc[i];
    }
}

// Final softmax over d of the refined cost -> norm_cost buffer [B,32,H,W]
__global__ void softmax_kernel(const float* __restrict__ cost, float* __restrict__ nc) {
    const int h = blockIdx.x, b = blockIdx.y, w = (int)threadIdx.x;
    float v[Dd];
    float mx = -1e30f;
    #pragma unroll
    for (int d = 0; d < Dd; ++d) {
        v[d] = cost[(((size_t)b * Dd + d) * Hh + h) * Ww + w];
        mx = fmaxf(mx, v[d]);
    }
    float sum = 0.f;
    #pragma unroll
    for (int d = 0; d < Dd; ++d) { v[d] = expf(v[d] - mx); sum += v[d]; }
    const float inv = 1.f / sum;
    #pragma unroll
    for (int d = 0; d < Dd; ++d)
        nc[(((size_t)b * Dd + d) * Hh + h) * Ww + w] = v[d] * inv;
}

// ---------------------------------------------------------------------------
extern "C" void kernel_launch(void* const* d_in, const int* in_sizes, int n_in,
                              void* d_out, int out_size, void* d_ws, size_t ws_size,
                              hipStream_t stream) {
    (void)in_sizes; (void)n_in; (void)out_size; (void)ws_size;
    const float* t_feat  = (const float*)d_in[0];
    const float* s_feat  = (const float*)d_in[1];
    const float* directs = (const float*)d_in[2];
    const float* qw = (const float*)d_in[3];
    const float* qb = (const float*)d_in[4];
    const float* kw = (const float*)d_in[5];
    const float* kb = (const float*)d_in[6];
    const float* vw = (const float*)d_in[7];
    const float* vb = (const float*)d_in[8];
    const float* p1w = (const float*)d_in[9];
    const float* p1b = (const float*)d_in[10];
    const float* p2w = (const float*)d_in[11];
    const float* p2b = (const float*)d_in[12];
    const float* rw = (const float*)d_in[13];
    const float* rb = (const float*)d_in[14];
    const int*   imgw = (const int*)d_in[15];

    float* out_x    = (float*)d_out;                        // [B,256,H,W]
    float* out_cost = out_x + (size_t)Bq * Cc * HW;         // [B,32,H,W]

    float* ws    = (float*)d_ws;
    float* qagg  = ws;                                      // [B,128,H,W] q | agg
    float* kbuf  = qagg + (size_t)Bq * 128 * HW;            // [B,64,H,W]
    float* vbuf  = kbuf + (size_t)Bq * 64 * HW;             // [B,64,H,W]
    float* cost0 = vbuf + (size_t)Bq * 64 * HW;             // [B,32,H,W]
    float* h1    = cost0 + (size_t)Bq * 32 * HW;            // [B,64,H,W]
    float* ncb   = h1 + (size_t)Bq * 64 * HW;               // [B,32,H,W]

    const dim3 gQKV(HW / 16 / NWAVE, CQ / 16, Bq);          // (192, 4, 2)
    gemm1x1_wmma<<<gQKV, 32 * NWAVE, 0, stream>>>(qw, qb, t_feat, Cc, (long)Cc * HW,
                                                  nullptr, 0, 0L, qagg, (long)128 * HW, CQ, 0);
    gemm1x1_wmma<<<gQKV, 32 * NWAVE, 0, stream>>>(kw, kb, s_feat, Cc, (long)Cc * HW,
                                                  nullptr, 0, 0L, kbuf, (long)CQ * HW, CQ, 0);
    gemm1x1_wmma<<<gQKV, 32 * NWAVE, 0, stream>>>(vw, vb, s_feat, Cc, (long)Cc * HW,
                                                  nullptr, 0, 0L, vbuf, (long)CQ * HW, CQ, 0);

    cost_kernel<<<dim3(Hh, Bq), dim3(Ww, 2), 0, stream>>>(qagg, kbuf, directs, imgw, cost0);
    softmax_agg_kernel<<<dim3(Hh, Bq), dim3(Ww, 2), 0, stream>>>(cost0, vbuf, directs,
                                                                 imgw, qagg);

    conv3x3_wmma<<<dim3(3, 64 / 16, Bq * Hh), 32 * NWAVE, 0, stream>>>(
        p1w, p1b, qagg, 128, h1, 64, nullptr, 0);
    conv3x3_wmma<<<dim3(3, 32 / 16, Bq * Hh), 32 * NWAVE, 0, stream>>>(
        p2w, p2b, h1, 64, out_cost, 32, cost0, 1);

    softmax_kernel<<<dim3(Hh, Bq), Ww, 0, stream>>>(out_cost, ncb);

    const dim3 gOut(HW / 16 / NWAVE, Cc / 16, Bq);          // (192, 16, 2)
    gemm1x1_wmma<<<gOut, 32 * NWAVE, 0, stream>>>(rw, rb, t_feat, Cc, (long)Cc * HW,
                                                  ncb, Dd, (long)Dd * HW, out_x,
                                                  (long)Cc * HW, Cc, 1);
}